// MvdrBeamformer_30769145709364
// MI455X (gfx1250) — compile-verified
//
#include <hip/hip_runtime.h>
#include <cstddef>

typedef float v2f __attribute__((ext_vector_type(2)));
typedef float v8f __attribute__((ext_vector_type(8)));

static constexpr int kN = 8, kC = 8, kF = 257, kT = 1500, kA = 512;
static constexpr int kFP = (kF + 1) / 2;  // 129 frequency pairs
static constexpr float kEps = 1.1920928955078125e-07f;  // FLT_EPSILON

// ---- workspace layout (in floats) ----
static constexpr size_t OFF_MAXABS = 0;                                  // N*F
static constexpr size_t OFF_MST    = OFF_MAXABS + (size_t)kN * kF;       // N*F*T
static constexpr size_t OFF_G      = OFF_MST + (size_t)kN * kF * kT;     // N*C
static constexpr size_t OFF_RSRE   = OFF_G + (size_t)kN * kC;            // N*F*64 each
static constexpr size_t OFF_RSIM   = OFF_RSRE + (size_t)kN * kF * 64;
static constexpr size_t OFF_RNRE   = OFF_RSIM + (size_t)kN * kF * 64;
static constexpr size_t OFF_RNIM   = OFF_RNRE + (size_t)kN * kF * 64;
static constexpr size_t OFF_WRE    = OFF_RNIM + (size_t)kN * kF * 64;    // N*F*C
static constexpr size_t OFF_WIM    = OFF_WRE + (size_t)kN * kF * kC;

static __device__ __forceinline__ v8f wmma4(v2f a, v2f b, v8f c) {
  // D = A(16x4 f32) * B(4x16 f32) + C(16x16 f32)
  return __builtin_amdgcn_wmma_f32_16x16x4_f32(false, a, false, b, (short)0, c,
                                               false, false);
}

// ---------------- K1a: per-(n,f) inf-norm of mask over time ----------------
__global__ void k_mask_max(const float* __restrict__ m, float* __restrict__ ws) {
  const int n = blockIdx.x;
  const int f = threadIdx.x;
  if (f >= kF) return;
  const float* p = m + (size_t)n * kT * kF + f;
  float mx = 0.0f;
  for (int t = 0; t < kT; ++t) mx = fmaxf(mx, fabsf(p[(size_t)t * kF]));
  ws[OFF_MAXABS + (size_t)n * kF + f] = mx;
}

// ------------- K1b: normalize + transpose mask to msT[n][f][t] -------------
__global__ void k_mask_transpose(const float* __restrict__ m,
                                 float* __restrict__ ws) {
  __shared__ float tile[32][33];
  const int n = blockIdx.z;
  const int f0 = blockIdx.x * 32;
  const int t0 = blockIdx.y * 32;
  const int tx = threadIdx.x, ty = threadIdx.y;  // blockDim = (32, 8)
#pragma unroll
  for (int r = 0; r < 4; ++r) {
    const int t = t0 + ty + r * 8, f = f0 + tx;
    float v = 0.0f;
    if (t < kT && f < kF) v = m[(size_t)n * kT * kF + (size_t)t * kF + f];
    tile[ty + r * 8][tx] = v;
  }
  __syncthreads();
#pragma unroll
  for (int r = 0; r < 4; ++r) {
    const int fl = ty + r * 8;
    const int f = f0 + fl, t = t0 + tx;
    if (t < kT && f < kF) {
      const float ma = ws[OFF_MAXABS + (size_t)n * kF + f];
      ws[OFF_MST + ((size_t)n * kF + f) * kT + t] = tile[tx][fl] / (ma + kEps);
    }
  }
}

// ---- K2: masked + full covariance via V_WMMA_F32_16X16X4_F32 ----
// Tile: rows/cols 0..7 -> (f0, c), 8..15 -> (f1, c). Diagonal 8x8 blocks kept.
__global__ void __launch_bounds__(128) k_cov(const float* __restrict__ xre,
                                             const float* __restrict__ xim,
                                             float* __restrict__ ws) {
  const int fp = blockIdx.x;
  const int n = blockIdx.y;
  const int tid = threadIdx.x;
  const int wave = tid >> 5;
  const int lane = tid & 31;
  const int row = lane & 15;  // M index (A) == N index (B) for this lane
  const int hi = lane >> 4;   // K half: lanes 16..31 carry K=2,3
  const int c = row & 7;
  const int fl = row >> 3;
  int f = 2 * fp + fl;
  if (f >= kF) f = kF - 1;  // clamp for the odd last pair; not stored

  const float* xr = xre + (((size_t)n * kC + c) * kF + f) * kT;
  const float* xi = xim + (((size_t)n * kC + c) * kF + f) * kT;
  const float* ms = ws + OFF_MST + ((size_t)n * kF + f) * kT;

  v8f aRsRe = {0, 0, 0, 0, 0, 0, 0, 0};
  v8f aP = aRsRe, aQ = aRsRe, aFRe = aRsRe, aPf = aRsRe, aQf = aRsRe;
  float msum = 0.0f;

  for (int j = wave; j < kT / 4; j += 4) {  // 375 K-chunks of 4 time samples
    const int to = 4 * j + 2 * hi;
    const v2f sre = *(const v2f*)(xr + to);
    const v2f sim = *(const v2f*)(xi + to);
    const v2f mv = *(const v2f*)(ms + to);
    msum += mv.x + mv.y;
    const v2f amre = mv * sre;
    const v2f amim = mv * sim;
    // R = sum_t (m*S)[.,t] * conj(S)[.,t]^T ; im split as P - Q (no A-neg on f32 WMMA)
    aRsRe = wmma4(amre, sre, aRsRe);
    aRsRe = wmma4(amim, sim, aRsRe);
    aP = wmma4(amim, sre, aP);
    aQ = wmma4(amre, sim, aQ);
    aFRe = wmma4(sre, sre, aFRe);
    aFRe = wmma4(sim, sim, aFRe);
    aPf = wmma4(sim, sre, aPf);
    aQf = wmma4(sre, sim, aQf);
  }

  __shared__ float red[4][6 * 256];
  __shared__ float fin[6 * 256];
  __shared__ float msums[128];
  __shared__ float ssum[2];

#pragma unroll
  for (int v = 0; v < 8; ++v) {
    red[wave][0 * 256 + v * 32 + lane] = aRsRe[v];
    red[wave][1 * 256 + v * 32 + lane] = aP[v];
    red[wave][2 * 256 + v * 32 + lane] = aQ[v];
    red[wave][3 * 256 + v * 32 + lane] = aFRe[v];
    red[wave][4 * 256 + v * 32 + lane] = aPf[v];
    red[wave][5 * 256 + v * 32 + lane] = aQf[v];
  }
  msums[tid] = msum;
  __syncthreads();
  for (int e = tid; e < 6 * 256; e += 128)
    fin[e] = red[0][e] + red[1][e] + red[2][e] + red[3][e];
  if (tid < 2) {  // mask sums per local frequency (each ms value seen by 8 c-lanes)
    float s = 0.0f;
    for (int i = 0; i < 128; ++i)
      if (((i >> 3) & 1) == tid) s += msums[i];
    ssum[tid] = s * 0.125f;
  }
  __syncthreads();

  // finalize: 128 threads -> (fl, c, d)
  const int flq = tid >> 6;
  const int cc = (tid >> 3) & 7;
  const int dd = tid & 7;
  const int fo = 2 * fp + flq;
  if (fo < kF) {
    const int M = 8 * flq + cc, Nn = 8 * flq + dd;
    const int v = (M < 8) ? M : (M - 8);
    const int ln = Nn + ((M < 8) ? 0 : 16);
    const int el = v * 32 + ln;  // C/D tile layout: VGPR v, lane ln
    const float rsre = fin[0 * 256 + el];
    const float rsim = fin[1 * 256 + el] - fin[2 * 256 + el];
    const float fre = fin[3 * 256 + el];
    const float fim = fin[4 * 256 + el] - fin[5 * 256 + el];
    const float sm = ssum[flq];
    const float dsv = fmaxf(sm, kEps);
    const float dnv = fmaxf((float)kT - sm, kEps);
    const size_t idx = (((size_t)n * kF + fo) * 8 + cc) * 8 + dd;
    ws[OFF_RSRE + idx] = rsre / dsv;
    ws[OFF_RSIM + idx] = rsim / dsv;
    ws[OFF_RNRE + idx] = (fre - rsre) / dnv + ((cc == dd) ? kEps : 0.0f);
    ws[OFF_RNIM + idx] = (fim - rsim) / dnv;
  }
}

// ---------------- K3: channel-attention logits g[n][c] ----------------
__global__ void __launch_bounds__(256) k_attn(const float* __restrict__ projw,
                                              const float* __restrict__ projb,
                                              const float* __restrict__ gw,
                                              const float* __restrict__ gb,
                                              float* __restrict__ ws) {
  const int n = blockIdx.x >> 3;
  const int c = blockIdx.x & 7;
  const int tid = threadIdx.x;
  __shared__ float featv[kF];
  __shared__ float rbuf[256];
  for (int f = tid; f < kF; f += 256) {
    float sre = 0.0f, sim = 0.0f;
    const size_t base = (((size_t)n * kF + f) * 8 + c) * 8;
#pragma unroll
    for (int d = 0; d < 8; ++d) {
      if (d == c) continue;
      sre += ws[OFF_RSRE + base + d];
      sim += ws[OFF_RSIM + base + d];
    }
    featv[f] = sqrtf(sre * sre + sim * sim) * (1.0f / 7.0f);
  }
  __syncthreads();
  float gp = 0.0f;
  for (int a = tid; a < kA; a += 256) {
    float dot = projb[a];
    const float* wrow = projw + (size_t)a * kF;
    for (int f = 0; f < kF; ++f) dot += featv[f] * wrow[f];
    gp += tanhf(dot) * gw[a];
  }
  rbuf[tid] = gp;
  __syncthreads();
  for (int s = 128; s > 0; s >>= 1) {
    if (tid < s) rbuf[tid] += rbuf[tid + s];
    __syncthreads();
  }
  if (tid == 0) ws[OFF_G + n * 8 + c] = rbuf[0] + gb[0];
}

// ---- K4: per-(n,f) complex Gauss-Jordan solve [Rn | Rs] -> X, w = X u / tr ----
// 8 lanes per problem; lane j owns column j of both halves.
__global__ void __launch_bounds__(64) k_solve(float* __restrict__ ws) {
  const int tid = threadIdx.x;
  const int j = tid & 7;
  const int p = blockIdx.x * 8 + (tid >> 3);
  const bool ok = p < kN * kF;
  const int pp = ok ? p : 0;
  const int n = pp / kF, f = pp % kF;

  // softmax over channels (logits global per n)
  float gm = -1e30f, gv;
  float se = 0.0f;
#pragma unroll
  for (int i = 0; i < 8; ++i) gm = fmaxf(gm, ws[OFF_G + n * 8 + i]);
#pragma unroll
  for (int i = 0; i < 8; ++i) se += expf(ws[OFF_G + n * 8 + i] - gm);
  gv = ws[OFF_G + n * 8 + j];
  const float myu = expf(gv - gm) / se;

  float Are[8], Aim[8], Bre[8], Bim[8];
  const size_t mbase = ((size_t)n * kF + f) * 64;
#pragma unroll
  for (int i = 0; i < 8; ++i) {
    const size_t idx = mbase + (size_t)i * 8 + j;
    Are[i] = ws[OFF_RNRE + idx];
    Aim[i] = ws[OFF_RNIM + idx];
    Bre[i] = ws[OFF_RSRE + idx];
    Bim[i] = ws[OFF_RSIM + idx];
  }

#pragma unroll
  for (int k = 0; k < 8; ++k) {
    const float pre = __shfl(Are[k], k, 8);
    const float pim = __shfl(Aim[k], k, 8);
    const float inv = 1.0f / (pre * pre + pim * pim);
    const float ire = pre * inv, iim = -pim * inv;
    const float tAre = Are[k] * ire - Aim[k] * iim;
    const float tAim = Are[k] * iim + Aim[k] * ire;
    const float tBre = Bre[k] * ire - Bim[k] * iim;
    const float tBim = Bre[k] * iim + Bim[k] * ire;
    Are[k] = tAre; Aim[k] = tAim; Bre[k] = tBre; Bim[k] = tBim;
    float fre[8], fim[8];
#pragma unroll
    for (int i = 0; i < 8; ++i) {
      fre[i] = __shfl(Are[i], k, 8);
      fim[i] = __shfl(Aim[i], k, 8);
    }
#pragma unroll
    for (int i = 0; i < 8; ++i) {
      if (i == k) continue;
      Are[i] -= fre[i] * tAre - fim[i] * tAim;
      Aim[i] -= fre[i] * tAim + fim[i] * tAre;
      Bre[i] -= fre[i] * tBre - fim[i] * tBim;
      Bim[i] -= fre[i] * tBim + fim[i] * tBre;
    }
  }

  // trace(X): lane j holds diagonal element X[j][j]
  float dre = 0.0f, dim_ = 0.0f;
#pragma unroll
  for (int i = 0; i < 8; ++i)
    if (i == j) { dre = Bre[i]; dim_ = Bim[i]; }
#pragma unroll
  for (int s = 1; s < 8; s <<= 1) {
    dre += __shfl_xor(dre, s, 8);
    dim_ += __shfl_xor(dim_, s, 8);
  }
  const float tre = dre + kEps, tim = dim_;

  // w_num = X u  (lane j contributes X[:,j] * u[j]); reduce across group
  float wr[8], wi[8];
#pragma unroll
  for (int i = 0; i < 8; ++i) { wr[i] = Bre[i] * myu; wi[i] = Bim[i] * myu; }
#pragma unroll
  for (int s = 1; s < 8; s <<= 1) {
#pragma unroll
    for (int i = 0; i < 8; ++i) {
      wr[i] += __shfl_xor(wr[i], s, 8);
      wi[i] += __shfl_xor(wi[i], s, 8);
    }
  }
  const float tinv = 1.0f / (tre * tre + tim * tim);
  float ore = 0.0f, oim = 0.0f;
#pragma unroll
  for (int i = 0; i < 8; ++i)
    if (i == j) { ore = wr[i]; oim = wi[i]; }
  const float wre = (ore * tre + oim * tim) * tinv;
  const float wim = (oim * tre - ore * tim) * tinv;
  if (ok) {
    ws[OFF_WRE + (size_t)p * 8 + j] = wre;
    ws[OFF_WIM + (size_t)p * 8 + j] = -wim;  // store conj(w) for beamforming
  }
}

// ---------------- K5: beamform + layout to N x T x F x 2 ----------------
__global__ void __launch_bounds__(256) k_beam(const float* __restrict__ xre,
                                              const float* __restrict__ xim,
                                              const float* __restrict__ ws,
                                              float* __restrict__ out) {
  const int p = blockIdx.x;
  const int n = p / kF, f = p % kF;
  const int tid = threadIdx.x;
  __shared__ float wr[8], wi[8];
  if (tid < 8) {
    wr[tid] = ws[OFF_WRE + (size_t)p * 8 + tid];
    wi[tid] = ws[OFF_WIM + (size_t)p * 8 + tid];
  }
  __syncthreads();
  const size_t rbase = ((size_t)n * kC * kF + f) * kT;
  for (int t = tid; t < kT; t += 256) {
    float ar = 0.0f, ai = 0.0f;
#pragma unroll
    for (int c = 0; c < 8; ++c) {
      const size_t o = rbase + (size_t)c * kF * kT + t;
      const float xr = xre[o], xiv = xim[o];
      ar += wr[c] * xr - wi[c] * xiv;  // w already conjugated
      ai += wr[c] * xiv + wi[c] * xr;
    }
    const size_t oo = (((size_t)n * kT + t) * kF + f) * 2;
    out[oo] = ar;
    out[oo + 1] = ai;
  }
}

extern "C" void kernel_launch(void* const* d_in, const int* in_sizes, int n_in,
                              void* d_out, int out_size, void* d_ws,
                              size_t ws_size, hipStream_t stream) {
  (void)in_sizes; (void)n_in; (void)out_size; (void)ws_size;
  const float* m      = (const float*)d_in[0];
  const float* x_re   = (const float*)d_in[1];
  const float* x_im   = (const float*)d_in[2];
  const float* proj_w = (const float*)d_in[3];
  const float* proj_b = (const float*)d_in[4];
  const float* gvec_w = (const float*)d_in[5];
  const float* gvec_b = (const float*)d_in[6];
  float* out = (float*)d_out;
  float* ws = (float*)d_ws;

  k_mask_max<<<kN, 288, 0, stream>>>(m, ws);
  k_mask_transpose<<<dim3((kF + 31) / 32, (kT + 31) / 32, kN), dim3(32, 8), 0,
                     stream>>>(m, ws);
  k_cov<<<dim3(kFP, kN), 128, 0, stream>>>(x_re, x_im, ws);
  k_attn<<<kN * kC, 256, 0, stream>>>(proj_w, proj_b, gvec_w, gvec_b, ws);
  k_solve<<<(kN * kF + 7) / 8, 64, 0, stream>>>(ws);
  k_beam<<<kN * kF, 256, 0, stream>>>(x_re, x_im, ws, out);
}